// LuongAttnDecoderRNNFeature_62319975464973
// MI455X (gfx1250) — compile-verified
//
#include <hip/hip_runtime.h>
#include <hip/hip_bf16.h>
#include <math.h>

typedef __attribute__((ext_vector_type(2))) float v2f;
typedef __attribute__((ext_vector_type(4))) float v4f;
typedef __attribute__((ext_vector_type(8))) float v8f;

#define Hh   1024
#define Ff   2
#define VOC  32000
#define Bb   64
#define Ll   1024
#define GIN  2052   // 2*(H+F)
#define LSPLIT 8    // attention L-partials per batch element

// ---------------------------------------------------------------------------
// Kernel 1: gather GRU input  x[b, 0:1024]=emb[tok], [1024:1026]=feature,
//           [1026:2052]=pos_emb[pos]
// ---------------------------------------------------------------------------
__global__ void build_x_kernel(const int* __restrict__ input_seq,
                               const int* __restrict__ positions,
                               const float* __restrict__ feature,
                               const float* __restrict__ emb,
                               const float* __restrict__ pos_emb,
                               float* __restrict__ x) {
    const int b = blockIdx.x;
    const int tok = input_seq[b];
    const int pos = positions[b];
    for (int i = threadIdx.x; i < GIN; i += blockDim.x) {
        float v;
        if (i < Hh)            v = emb[(size_t)tok * Hh + i];
        else if (i < Hh + Ff)  v = feature[b * Ff + (i - Hh)];
        else                   v = pos_emb[(size_t)pos * (Hh + Ff) + (i - (Hh + Ff))];
        x[(size_t)b * GIN + i] = v;
    }
}

// ---------------------------------------------------------------------------
// Kernel 2 (generic): out[64, N] = act(A[64,K] @ W[N,K]^T + bias)
// WMMA f32 16x16x4. One block covers all of M=64 (W read exactly once).
// 8 waves/block, each wave = one 16-wide N-tile with 4 M-accumulators.
// KC=64 K-chunks staged in LDS; inner body fully unrolled: 16 hoisted
// non-temporal B-fragment loads feeding 64 WMMAs per chunk.
// ---------------------------------------------------------------------------
#define KC 64
#define LDSW (KC + 4)
__global__ void wmma_gemm_f32_kernel(const float* __restrict__ A,
                                     const float* __restrict__ W,
                                     const float* __restrict__ bias,
                                     float* __restrict__ out,
                                     int N, int K, int ldo, int act) {
    __shared__ float As[64 * LDSW];
    const int tid   = threadIdx.x;
    const int wave  = tid >> 5;
    const int lane  = tid & 31;
    const int n0    = (blockIdx.x * 8 + wave) * 16;
    const int lrow  = lane & 15;        // N index (B) / M index (A) within tile
    const int khalf = (lane >> 4) * 2;  // K sub-offset per lane half

    v8f acc[4] = {};
    const float* wrow = W + (size_t)(n0 + lrow) * K + khalf;

    for (int k0 = 0; k0 < K; k0 += KC) {
        const int kc = min(KC, K - k0);
        __syncthreads();
        for (int i = tid; i < 64 * kc; i += 256) {
            const int r = i / kc;
            const int c = i - r * kc;
            As[r * LDSW + c] = A[(size_t)r * K + k0 + c];
        }
        __syncthreads();
        const float* wk = wrow + k0;
        if (kc == KC) {
            v2f bf[16];
#pragma unroll
            for (int s = 0; s < 16; ++s)
                bf[s] = __builtin_nontemporal_load((const v2f*)(wk + 4 * s));
#pragma unroll
            for (int s = 0; s < 16; ++s) {
                const int kk = 4 * s;
#pragma unroll
                for (int t = 0; t < 4; ++t) {
                    v2f a = *(const v2f*)(&As[(16 * t + lrow) * LDSW + khalf + kk]);
                    acc[t] = __builtin_amdgcn_wmma_f32_16x16x4_f32(
                        false, a, false, bf[s], (short)0, acc[t], false, false);
                }
            }
        } else {
            for (int kk = 0; kk < kc; kk += 4) {
                v2f b = __builtin_nontemporal_load((const v2f*)(wk + kk));
#pragma unroll
                for (int t = 0; t < 4; ++t) {
                    v2f a = *(const v2f*)(&As[(16 * t + lrow) * LDSW + khalf + kk]);
                    acc[t] = __builtin_amdgcn_wmma_f32_16x16x4_f32(
                        false, a, false, b, (short)0, acc[t], false, false);
                }
            }
        }
    }

    // D layout: element r at lane L -> M = r + 8*(L>>4), N = L&15
    const int rbase = (lane >> 4) * 8;
    const int nn = n0 + lrow;
    const float bv = bias[nn];
#pragma unroll
    for (int t = 0; t < 4; ++t) {
#pragma unroll
        for (int r = 0; r < 8; ++r) {
            float v = acc[t][r] + bv;
            if (act) v = tanhf(v);
            out[(size_t)(16 * t + rbase + r) * ldo + nn] = v;
        }
    }
}

// ---------------------------------------------------------------------------
// Kernel 3: GRU gates (torch order r,z,n).  Writes h to scratch, to the
// [h|context] concat buffer, and to the h section of d_out.
// ---------------------------------------------------------------------------
__global__ void gru_gate_kernel(const float* __restrict__ gx,
                                const float* __restrict__ gh,
                                const float* __restrict__ h0,
                                float* __restrict__ h_ws,
                                float* __restrict__ hc,
                                float* __restrict__ h_out) {
    const int idx = blockIdx.x * blockDim.x + threadIdx.x;   // B*H threads
    const int b = idx >> 10;
    const int i = idx & (Hh - 1);
    const size_t g = (size_t)b * 3 * Hh + i;
    const float xr = gx[g], xz = gx[g + Hh], xn = gx[g + 2 * Hh];
    const float hr = gh[g], hz = gh[g + Hh], hn = gh[g + 2 * Hh];
    const float r = 1.0f / (1.0f + __expf(-(xr + hr)));
    const float z = 1.0f / (1.0f + __expf(-(xz + hz)));
    const float n = tanhf(xn + r * hn);
    const float hv = (1.0f - z) * n + z * h0[idx];
    h_ws[idx] = hv;
    hc[(size_t)b * 2 * Hh + i] = hv;
    h_out[idx] = hv;
}

// ---------------------------------------------------------------------------
// Kernel 4a: partial fused attention. grid = (B, LSPLIT); each block streams
// L/LSPLIT = 128 rows of encoder_outputs once (non-temporal), maintaining
// online max/sum and an unnormalized running context; raw energies + (m,s)
// + partial context go to workspace.
// ---------------------------------------------------------------------------
__global__ void attention_partial_kernel(const float* __restrict__ enc, // [L,B,H]
                                         const float* __restrict__ h,   // [B,H]
                                         float* __restrict__ eWs,       // [B,L]
                                         float* __restrict__ ctxP,      // [LSPLIT,B,H]
                                         float* __restrict__ msP) {     // [LSPLIT,B,2]
    __shared__ float hS[Hh];
    __shared__ float encT[8][Hh];
    __shared__ float eChunk[8];
    const int b    = blockIdx.x;
    const int part = blockIdx.y;
    const int tid  = threadIdx.x;
    const int wave = tid >> 5;
    const int lane = tid & 31;
    const int lbase = part * (Ll / LSPLIT);

    for (int i = tid; i < Hh; i += 256) hS[i] = h[(size_t)b * Hh + i];
    __syncthreads();

    float m = -INFINITY, s = 0.0f;
    float c0 = 0.0f, c1 = 0.0f, c2 = 0.0f, c3 = 0.0f;  // ctx cols tid*4..+3
    const int col0 = tid * 4;

    for (int c = 0; c < (Ll / LSPLIT) / 8; ++c) {
        const int l = lbase + c * 8 + wave;
        const float* row = enc + ((size_t)l * Bb + b) * Hh;
        float dot = 0.0f;
#pragma unroll
        for (int k = 0; k < 8; ++k) {
            const int cc = k * 128 + lane * 4;
            const v4f v = __builtin_nontemporal_load((const v4f*)(row + cc));
            *(v4f*)(&encT[wave][cc]) = v;
            dot += v.x * hS[cc] + v.y * hS[cc + 1] +
                   v.z * hS[cc + 2] + v.w * hS[cc + 3];
        }
#pragma unroll
        for (int off = 16; off > 0; off >>= 1)
            dot += __shfl_down(dot, off, 32);
        if (lane == 0) {
            eChunk[wave] = dot;
            eWs[(size_t)b * Ll + l] = dot;
        }
        __syncthreads();

        float mc = m;
#pragma unroll
        for (int r = 0; r < 8; ++r) mc = fmaxf(mc, eChunk[r]);
        const float scale = __expf(m - mc);
        s *= scale; c0 *= scale; c1 *= scale; c2 *= scale; c3 *= scale;
#pragma unroll
        for (int r = 0; r < 8; ++r) {
            const float p = __expf(eChunk[r] - mc);
            s += p;
            c0 += p * encT[r][col0];
            c1 += p * encT[r][col0 + 1];
            c2 += p * encT[r][col0 + 2];
            c3 += p * encT[r][col0 + 3];
        }
        m = mc;
        __syncthreads();
    }

    float* cp = ctxP + ((size_t)part * Bb + b) * Hh;
    cp[col0]     = c0;
    cp[col0 + 1] = c1;
    cp[col0 + 2] = c2;
    cp[col0 + 3] = c3;
    if (tid == 0) {
        msP[((size_t)part * Bb + b) * 2]     = m;
        msP[((size_t)part * Bb + b) * 2 + 1] = s;
    }
}

// ---------------------------------------------------------------------------
// Kernel 4b: combine LSPLIT partials -> context (into [h|ctx] buffer) and
// normalized attn (into d_out). Deterministic fixed-order merge.
// ---------------------------------------------------------------------------
__global__ void attention_combine_kernel(const float* __restrict__ eWs,
                                         const float* __restrict__ ctxP,
                                         const float* __restrict__ msP,
                                         float* __restrict__ hc,        // ctx -> [:,H:]
                                         float* __restrict__ attn_out) {
    const int b = blockIdx.x;
    const int tid = threadIdx.x;

    float M = -INFINITY;
#pragma unroll
    for (int p = 0; p < LSPLIT; ++p)
        M = fmaxf(M, msP[((size_t)p * Bb + b) * 2]);
    float S = 0.0f;
    float w[LSPLIT];
#pragma unroll
    for (int p = 0; p < LSPLIT; ++p) {
        w[p] = __expf(msP[((size_t)p * Bb + b) * 2] - M);
        S += msP[((size_t)p * Bb + b) * 2 + 1] * w[p];
    }
    const float inv = 1.0f / S;

    const int col0 = tid * 4;
    float c0 = 0.0f, c1 = 0.0f, c2 = 0.0f, c3 = 0.0f;
#pragma unroll
    for (int p = 0; p < LSPLIT; ++p) {
        const float* cp = ctxP + ((size_t)p * Bb + b) * Hh;
        c0 += w[p] * cp[col0];
        c1 += w[p] * cp[col0 + 1];
        c2 += w[p] * cp[col0 + 2];
        c3 += w[p] * cp[col0 + 3];
    }
    float* ctx = hc + (size_t)b * 2 * Hh + Hh;
    ctx[col0]     = c0 * inv;
    ctx[col0 + 1] = c1 * inv;
    ctx[col0 + 2] = c2 * inv;
    ctx[col0 + 3] = c3 * inv;

    for (int l = tid; l < Ll; l += 256)
        attn_out[(size_t)b * Ll + l] = __expf(eWs[(size_t)b * Ll + l] - M) * inv;
}

// ---------------------------------------------------------------------------
extern "C" void kernel_launch(void* const* d_in, const int* in_sizes, int n_in,
                              void* d_out, int out_size, void* d_ws, size_t ws_size,
                              hipStream_t stream) {
    const int*   input_seq = (const int*)d_in[0];
    const float* last_hidden = (const float*)d_in[1];  // [1,B,H]
    const float* enc       = (const float*)d_in[2];    // [L,B,H]
    const int*   positions = (const int*)d_in[3];
    const float* feature   = (const float*)d_in[4];    // [B,F]
    const float* emb       = (const float*)d_in[5];    // [VOC,H]
    const float* pos_emb   = (const float*)d_in[6];    // [MAX_POS,H+F]
    const float* W_ih      = (const float*)d_in[7];    // [3H,GIN]
    const float* W_hh      = (const float*)d_in[8];    // [3H,H]
    const float* b_ih      = (const float*)d_in[9];
    const float* b_hh      = (const float*)d_in[10];
    const float* W_c       = (const float*)d_in[11];   // [H,2H]
    const float* b_c       = (const float*)d_in[12];
    const float* W_o       = (const float*)d_in[13];   // [VOC,H]
    const float* b_o       = (const float*)d_in[14];

    float* out      = (float*)d_out;                   // [B,VOC]
    float* h_out    = out + (size_t)Bb * VOC;          // [B,H]
    float* attn_out = h_out + (size_t)Bb * Hh;         // [B,L]

    float* ws   = (float*)d_ws;
    float* x    = ws;                            // [B,GIN]
    float* gx   = x    + (size_t)Bb * GIN;       // [B,3H]
    float* gh   = gx   + (size_t)Bb * 3 * Hh;    // [B,3H]
    float* hW   = gh   + (size_t)Bb * 3 * Hh;    // [B,H]
    float* hcW  = hW   + (size_t)Bb * Hh;        // [B,2H]
    float* coW  = hcW  + (size_t)Bb * 2 * Hh;    // [B,H]
    float* eWs  = coW  + (size_t)Bb * Hh;        // [B,L]
    float* ctxP = eWs  + (size_t)Bb * Ll;        // [LSPLIT,B,H]
    float* msP  = ctxP + (size_t)LSPLIT * Bb * Hh; // [LSPLIT,B,2]

    // 1. gather GRU input
    build_x_kernel<<<Bb, 256, 0, stream>>>(input_seq, positions, feature,
                                           emb, pos_emb, x);
    // 2. gx = x @ W_ih^T + b_ih   (N=3072, K=2052)
    wmma_gemm_f32_kernel<<<3 * Hh / 128, 256, 0, stream>>>(
        x, W_ih, b_ih, gx, 3 * Hh, GIN, 3 * Hh, 0);
    // 3. gh = h0 @ W_hh^T + b_hh  (N=3072, K=1024)
    wmma_gemm_f32_kernel<<<3 * Hh / 128, 256, 0, stream>>>(
        last_hidden, W_hh, b_hh, gh, 3 * Hh, Hh, 3 * Hh, 0);
    // 4. GRU gates -> h (scratch + concat buffer + d_out)
    gru_gate_kernel<<<(Bb * Hh) / 256, 256, 0, stream>>>(gx, gh, last_hidden,
                                                         hW, hcW, h_out);
    // 5a. fused attention partials: single NT pass over encoder_outputs
    attention_partial_kernel<<<dim3(Bb, LSPLIT), 256, 0, stream>>>(
        enc, hW, eWs, ctxP, msP);
    // 5b. merge partials -> context + attn
    attention_combine_kernel<<<Bb, 256, 0, stream>>>(eWs, ctxP, msP,
                                                     hcW, attn_out);
    // 6. concat_out = tanh([h|ctx] @ W_c^T + b_c)  (N=1024, K=2048)
    wmma_gemm_f32_kernel<<<Hh / 128, 256, 0, stream>>>(
        hcW, W_c, b_c, coW, Hh, 2 * Hh, Hh, 1);
    // 7. output = concat_out @ W_o^T + b_o  (N=32000, K=1024)
    wmma_gemm_f32_kernel<<<VOC / 128, 256, 0, stream>>>(
        coW, W_o, b_o, out, VOC, Hh, VOC, 0);
}